// TaskSpan1_33861522162529
// MI455X (gfx1250) — compile-verified
//
#include <hip/hip_runtime.h>

// ---------------------------------------------------------------------------
// Task: windowed span attention + MLP head (fp32)
//   B=2, L=1024, D=256, W=10, H_ATTN=128, H_NET=512, N_LABELS=17
// Dominant GEMM: M=20480, K=256, N=512 (5.4 GFLOP fp32) -> V_WMMA_F32_16X16X4_F32
// 32-row M-tiles halve net_w L2 traffic vs 16-row tiles; A tile staged to LDS
// via GLOBAL_LOAD_ASYNC_TO_LDS_B128 (ASYNCcnt) when the builtin is available.
// ---------------------------------------------------------------------------

typedef __attribute__((ext_vector_type(2))) float v2f;
typedef __attribute__((ext_vector_type(8))) float v8f;
typedef __attribute__((ext_vector_type(4))) int   v4i;

typedef __attribute__((address_space(1))) v4i* g_v4i_ptr;  // global
typedef __attribute__((address_space(3))) v4i* l_v4i_ptr;  // LDS

#define WMMA_F32X4(a, b, c) \
  __builtin_amdgcn_wmma_f32_16x16x4_f32(false, (a), false, (b), (short)0, (c), false, false)

#if defined(__has_builtin)
#if __has_builtin(__builtin_amdgcn_global_load_async_to_lds_b128)
#define HAVE_ASYNC_LDS 1
#endif
#if __has_builtin(__builtin_amdgcn_s_wait_asynccnt)
#define HAVE_WAIT_ASYNC_BUILTIN 1
#endif
#endif

__device__ __forceinline__ void wait_asynccnt0() {
#if defined(HAVE_WAIT_ASYNC_BUILTIN)
  __builtin_amdgcn_s_wait_asynccnt(0);
#else
  asm volatile("s_wait_asynccnt 0x0" ::: "memory");
#endif
}

constexpr int Bc = 2, Lc = 1024, Dc = 256, Wc = 10;
constexpr int HA = 128, HN = 512, NL = 17;
constexpr int M1 = Bc * Lc;        // 2048 scored positions
constexpr int M3 = Bc * Lc * Wc;   // 20480 span rows
constexpr int MT = 32;             // head-GEMM M tile (2 x 16-row WMMA subtiles)

// ---------------------------------------------------------------------------
// Kernel 1: per-position attention score
//   score[b,p] = relu(x_p @ W1 + b1) @ w2 + b2       (16 pos x 128 H per block)
// 4 waves/block, each wave owns 2 N-tiles of H_ATTN. WMMA C-layout:
//   VGPR r, lane j<16 -> (m=r, n=n0+j); lane j+16 -> (m=r+8, n=n0+j)
// ---------------------------------------------------------------------------
__global__ __launch_bounds__(128) void score_kernel(
    const float* __restrict__ x, const float* __restrict__ w1,
    const float* __restrict__ b1, const float* __restrict__ w2,
    const float* __restrict__ b2, float* __restrict__ scores) {
  __shared__ float sc[16];
  const int tid  = threadIdx.x;
  const int wave = tid >> 5;
  const int lane = tid & 31;
  const int half = lane >> 4;     // wave32: halves of the C/A/B layouts
  const int j    = lane & 15;
  const int p0   = blockIdx.x * 16;

  if (tid < 16) sc[tid] = 0.f;
  __syncthreads();

  const int n0 = wave * 32;
  const int n1 = n0 + 16;
  v8f c0 = {}, c1 = {};
  // A fragment base: row (p0+j), k = kk + 2*half + {0,1}
  const float* arow = x + (size_t)(p0 + j) * Dc + 2 * half;
#pragma unroll 4
  for (int kk = 0; kk < Dc; kk += 4) {
    v2f a = *(const v2f*)(arow + kk);
    const float* bp = w1 + (size_t)(kk + 2 * half) * HA + j;
    v2f bA, bB;
    bA.x = bp[n0];      bA.y = bp[HA + n0];
    bB.x = bp[n1];      bB.y = bp[HA + n1];
    c0 = WMMA_F32X4(a, bA, c0);
    c1 = WMMA_F32X4(a, bB, c1);
  }

  // fused epilogue: relu(+b1) * w2, reduce over N
  const float b1a = b1[n0 + j], b1b = b1[n1 + j];
  const float w2a = w2[n0 + j], w2b = w2[n1 + j];
  float part[8];
#pragma unroll
  for (int r = 0; r < 8; ++r) {
    float va = c0[r] + b1a; va = va > 0.f ? va : 0.f;
    float vb = c1[r] + b1b; vb = vb > 0.f ? vb : 0.f;
    part[r] = va * w2a + vb * w2b;
  }
  // reduce across the 16 lanes of each half (xor masks 1..8 never cross halves)
#pragma unroll
  for (int r = 0; r < 8; ++r) {
    float v = part[r];
    for (int m = 1; m < 16; m <<= 1) v += __shfl_xor(v, m, 32);
    part[r] = v;
  }
  if (j == 0) {
#pragma unroll
    for (int r = 0; r < 8; ++r) atomicAdd(&sc[half * 8 + r], part[r]);
  }
  __syncthreads();
  if (tid < 16) scores[p0 + tid] = sc[tid] + b2[0];
}

// ---------------------------------------------------------------------------
// Kernel 2: masked softmax over window + span vectors
//   span[b,l,v,:] = sum_{w<=v} softmax_w(score[b,clip(l+w)]) * x[b,clip(l+w),:]
// one block per (b,l); thread d in [0,256) owns feature d.
// ---------------------------------------------------------------------------
__global__ __launch_bounds__(256) void span_kernel(
    const float* __restrict__ x, const float* __restrict__ scores,
    float* __restrict__ span) {
  __shared__ float s[Wc];
  __shared__ float probs[Wc][Wc];
  const int bl  = blockIdx.x;         // b*L + l
  const int b   = bl / Lc;
  const int l   = bl - b * Lc;
  const int tid = threadIdx.x;

  if (tid < Wc) {
    int p = l + tid; if (p > Lc - 1) p = Lc - 1;
    s[tid] = scores[b * Lc + p];
  }
  __syncthreads();
  if (tid < Wc) {
    const int v = tid;
    float mx = -3.0e38f;
    for (int w = 0; w <= v; ++w) mx = fmaxf(mx, s[w]);
    float e[Wc], den = 0.f;
    for (int w = 0; w <= v; ++w) { e[w] = expf(s[w] - mx); den += e[w]; }
    const float inv = 1.f / den;
    for (int w = 0; w < Wc; ++w) probs[v][w] = (w <= v) ? e[w] * inv : 0.f;
  }
  __syncthreads();

  const int d = tid;                  // D == blockDim.x == 256
  float acc[Wc];
#pragma unroll
  for (int v = 0; v < Wc; ++v) acc[v] = 0.f;
#pragma unroll
  for (int w = 0; w < Wc; ++w) {
    int p = l + w; if (p > Lc - 1) p = Lc - 1;
    const float xv = x[((size_t)b * Lc + p) * Dc + d];
    for (int v = w; v < Wc; ++v) acc[v] = fmaf(probs[v][w], xv, acc[v]);
  }
  float* o = span + (size_t)bl * Wc * Dc + d;
#pragma unroll
  for (int v = 0; v < Wc; ++v) o[v * Dc] = acc[v];
}

// ---------------------------------------------------------------------------
// Kernel 3: head GEMM  hidden = relu(span @ net_w + net_b)  [32 x 512 tile]
//           logits = hidden @ out_w + out_b, seq-masked
// 8 waves/block; wave w owns hidden columns [64w, 64w+64) for BOTH 16-row
// M-subtiles, so every net_w fragment is reused across 32 rows. smem (64 KB)
// holds the 32x256 A tile during the k-loop, then is reused for the 32x512
// hidden tile (A is dead after the post-loop barrier).
// ---------------------------------------------------------------------------
__global__ __launch_bounds__(256) void head_kernel(
    const float* __restrict__ span, const float* __restrict__ netw,
    const float* __restrict__ netb, const float* __restrict__ outw,
    const float* __restrict__ outb, const int* __restrict__ seqlen,
    float* __restrict__ out) {
  __shared__ float smem[MT * HN];   // 64 KB: A tile (32x256) then hidden (32x512)
  const int tid  = threadIdx.x;
  const int wave = tid >> 5;
  const int lane = tid & 31;
  const int half = lane >> 4;
  const int j    = lane & 15;
  const int m0   = blockIdx.x * MT;

  if (tid == 0) __builtin_prefetch(outw, 0, 3);   // global_prefetch_b8

  // ---- stage A tile (32 x 256 f32 = 32 KB) into LDS ----
#if defined(HAVE_ASYNC_LDS)
  {
    const char* g = (const char*)(span + (size_t)m0 * Dc);
    char* l = (char*)smem;
    constexpr int bytes = MT * Dc * 4;
#pragma unroll
    for (int i = 0; i < bytes / (256 * 16); ++i) {
      const int off = (i * 256 + tid) * 16;
      __builtin_amdgcn_global_load_async_to_lds_b128(
          (g_v4i_ptr)(uintptr_t)(g + off),   // AS1 v4i* (inttoptr: global==generic)
          (l_v4i_ptr)(l + off),              // AS3 v4i* (addrspacecast)
          0, 0);
    }
    wait_asynccnt0();
  }
#else
  {
    const float4* src = (const float4*)(span + (size_t)m0 * Dc);
    float4* dst = (float4*)smem;
    for (int i = tid; i < MT * Dc / 4; i += 256) dst[i] = src[i];
  }
#endif
  __syncthreads();

  const int nbase = wave * 64;
  v8f acc0[4] = {};                 // M rows [0,16)
  v8f acc1[4] = {};                 // M rows [16,32)
  const float* arow0 = smem + (size_t)j * Dc + 2 * half;
  const float* arow1 = smem + (size_t)(16 + j) * Dc + 2 * half;
#pragma unroll 4
  for (int kk = 0; kk < Dc; kk += 4) {
    v2f a0 = *(const v2f*)(arow0 + kk);      // ds_load_b64
    v2f a1 = *(const v2f*)(arow1 + kk);
    const float* bp = netw + (size_t)(kk + 2 * half) * HN + nbase + j;
#pragma unroll
    for (int t = 0; t < 4; ++t) {
      v2f bb;
      bb.x = bp[t * 16];
      bb.y = bp[HN + t * 16];
      acc0[t] = WMMA_F32X4(a0, bb, acc0[t]); // B fragment reused across both
      acc1[t] = WMMA_F32X4(a1, bb, acc1[t]); // M-subtiles
    }
  }
  __syncthreads();                  // A tile dead; smem becomes hidden tile

  // relu + bias -> LDS hidden tile (32 x 512)
#pragma unroll
  for (int t = 0; t < 4; ++t) {
    const int n = nbase + t * 16 + j;
    const float bias = netb[n];
#pragma unroll
    for (int r = 0; r < 8; ++r) {
      float v0 = acc0[t][r] + bias; v0 = v0 > 0.f ? v0 : 0.f;
      float v1 = acc1[t][r] + bias; v1 = v1 > 0.f ? v1 : 0.f;
      smem[(half * 8 + r) * HN + n]        = v0;
      smem[(16 + half * 8 + r) * HN + n]   = v1;
    }
  }
  __syncthreads();

  // phase 2: 32 rows x 17 labels, K=512 from LDS; then sequence mask
  for (int o = tid; o < MT * NL; o += 256) {
    const int row = o / NL;
    const int col = o - row * NL;
    float a2 = outb[col];
    for (int n = 0; n < HN; ++n)
      a2 = fmaf(smem[row * HN + n], outw[n * NL + col], a2);
    const int gm  = m0 + row;                 // (b*L + l)*W + v
    const int b   = gm / (Lc * Wc);
    const int rem = gm - b * (Lc * Wc);
    const int l   = rem / Wc;
    const int v   = rem - l * Wc;
    const bool valid = (l + v) < seqlen[b];
    out[(size_t)gm * NL + col] = valid ? a2 : 0.f;
  }
}

// ---------------------------------------------------------------------------
// launch
// ---------------------------------------------------------------------------
extern "C" void kernel_launch(void* const* d_in, const int* in_sizes, int n_in,
                              void* d_out, int out_size, void* d_ws, size_t ws_size,
                              hipStream_t stream) {
  const float* inputs = (const float*)d_in[0];
  const int*   seqlen = (const int*)d_in[1];
  const float* w1     = (const float*)d_in[2];
  const float* b1     = (const float*)d_in[3];
  const float* w2     = (const float*)d_in[4];
  const float* b2     = (const float*)d_in[5];
  const float* netw   = (const float*)d_in[6];
  const float* netb   = (const float*)d_in[7];
  const float* outw   = (const float*)d_in[8];
  const float* outb   = (const float*)d_in[9];
  float* out = (float*)d_out;

  // workspace layout: [0, 2048) scores | [2048, 2048 + 20480*256) span_vec
  float* scores = (float*)d_ws;
  float* span   = scores + M1;

  score_kernel<<<M1 / 16, 128, 0, stream>>>(inputs, w1, b1, w2, b2, scores);
  span_kernel<<<Bc * Lc, 256, 0, stream>>>(inputs, scores, span);
  head_kernel<<<M3 / MT, 256, 0, stream>>>(span, netw, netb, outw, outb, seqlen, out);
}